// GCNBackbone_55817394979590
// MI455X (gfx1250) — compile-verified
//
#include <hip/hip_runtime.h>
#include <hip/hip_bf16.h>

typedef __attribute__((ext_vector_type(2))) float v2f;
typedef __attribute__((ext_vector_type(8))) float v8f;

#define D 64
#define NLAYERS 4
#define LN_EPS 1e-5f

// ---------------------------------------------------------------------------
// degree / dinv setup
// ---------------------------------------------------------------------------
__global__ void k_zero_f32(float* __restrict__ p, int n) {
    for (int i = blockIdx.x * blockDim.x + threadIdx.x; i < n; i += gridDim.x * blockDim.x)
        p[i] = 0.0f;
}

__global__ void k_count_deg(const int* __restrict__ dst, float* __restrict__ deg, int E) {
    for (int i = blockIdx.x * blockDim.x + threadIdx.x; i < E; i += gridDim.x * blockDim.x)
        atomicAdd(&deg[dst[i]], 1.0f);
}

__global__ void k_make_dinv(float* __restrict__ deg, int n) {
    for (int i = blockIdx.x * blockDim.x + threadIdx.x; i < n; i += gridDim.x * blockDim.x)
        deg[i] = rsqrtf(deg[i] + 1.0f);   // deg includes self-loop (+1)
}

__global__ void k_zero2(float* __restrict__ red) {
    red[0] = 0.0f;
    red[1] = 0.0f;
}

// ---------------------------------------------------------------------------
// GEMM: hw = h @ W   (fp32 WMMA 16x16x4), fused epilogue:
//   agg = dinv[row]^2 * hw + bias      (self-loop term + bias)
// One wave32 computes a 16-row x 64-col output tile. 4 waves / block.
// ---------------------------------------------------------------------------
__global__ void __launch_bounds__(128)
k_gemm_wmma(const float* __restrict__ h, const float* __restrict__ W,
            const float* __restrict__ bias, const float* __restrict__ dinv,
            float* __restrict__ hw, float* __restrict__ agg, int nrows) {
    const int wave = threadIdx.x >> 5;
    const int lane = threadIdx.x & 31;
    const int tile = blockIdx.x * 4 + wave;          // wave-uniform
    if (tile * 16 >= nrows) return;                  // uniform exit: EXEC all-1 for WMMA
    const int m0   = tile * 16;
    const int r    = lane & 15;
    const int half = lane >> 4;

    v8f acc0 = {}, acc1 = {}, acc2 = {}, acc3 = {};

    const float* arow = h + (size_t)(m0 + r) * D;
    #pragma unroll
    for (int k0 = 0; k0 < D; k0 += 4) {
        // A fragment (16x4 f32): lane<16 holds K=k0,k0+1 ; lane>=16 holds K=k0+2,k0+3
        const int ka = k0 + 2 * half;
        v2f a;
        a.x = arow[ka + 0];
        a.y = arow[ka + 1];
        // B fragments (4x16 f32) for 4 N-blocks
        const float* w0 = W + (size_t)ka * D;        // row k
        const float* w1 = w0 + D;                    // row k+1
        v2f b0, b1, b2, b3;
        b0.x = w0[ 0 + r]; b0.y = w1[ 0 + r];
        b1.x = w0[16 + r]; b1.y = w1[16 + r];
        b2.x = w0[32 + r]; b2.y = w1[32 + r];
        b3.x = w0[48 + r]; b3.y = w1[48 + r];
        acc0 = __builtin_amdgcn_wmma_f32_16x16x4_f32(false, a, false, b0, (short)0, acc0, false, false);
        acc1 = __builtin_amdgcn_wmma_f32_16x16x4_f32(false, a, false, b1, (short)0, acc1, false, false);
        acc2 = __builtin_amdgcn_wmma_f32_16x16x4_f32(false, a, false, b2, (short)0, acc2, false, false);
        acc3 = __builtin_amdgcn_wmma_f32_16x16x4_f32(false, a, false, b3, (short)0, acc3, false, false);
    }

    // C/D layout: VGPR j holds M = m0 + j + 8*half, N = r (+ n-block offset)
    #pragma unroll
    for (int j = 0; j < 8; ++j) {
        const int row = m0 + j + 8 * half;
        const float di = dinv[row];
        const float sn = di * di;
        float* hwr = hw  + (size_t)row * D;
        float* agr = agg + (size_t)row * D;
        const float v0 = acc0[j], v1 = acc1[j], v2 = acc2[j], v3 = acc3[j];
        hwr[ 0 + r] = v0;  agr[ 0 + r] = sn * v0 + bias[ 0 + r];
        hwr[16 + r] = v1;  agr[16 + r] = sn * v1 + bias[16 + r];
        hwr[32 + r] = v2;  agr[32 + r] = sn * v2 + bias[32 + r];
        hwr[48 + r] = v3;  agr[48 + r] = sn * v3 + bias[48 + r];
    }
}

// ---------------------------------------------------------------------------
// Edge scatter: agg[dst] += dinv[src]*dinv[dst] * hw[src]
// One wave32 per edge iteration: 64 channels -> float2 per lane.
// ---------------------------------------------------------------------------
__global__ void __launch_bounds__(256)
k_scatter_edges(const int* __restrict__ ei, const float* __restrict__ hw,
                const float* __restrict__ dinv, float* __restrict__ agg, int E) {
    const int lane  = threadIdx.x & 31;
    const int wid   = (blockIdx.x * blockDim.x + threadIdx.x) >> 5;
    const int nwave = (gridDim.x * blockDim.x) >> 5;
    for (int e = wid; e < E; e += nwave) {
        const int src = ei[e];
        const int dst = ei[E + e];
        // prefetch next iteration's source row (global_prefetch_b8)
        const int en = e + nwave;
        if (en < E)
            __builtin_prefetch(hw + (size_t)ei[en] * D + lane * 2, 0, 1);
        const float w = dinv[src] * dinv[dst];
        const float2 v = *(const float2*)(hw + (size_t)src * D + lane * 2);
        float* out = agg + (size_t)dst * D + lane * 2;
        atomicAdd(out + 0, w * v.x);
        atomicAdd(out + 1, w * v.y);
    }
}

// ---------------------------------------------------------------------------
// Graph-wide LayerNorm stats: sum and sum-of-squares over all N*D values.
// ---------------------------------------------------------------------------
__global__ void __launch_bounds__(256)
k_reduce_stats(const float* __restrict__ agg, float* __restrict__ red, int n) {
    __shared__ float s0[256];
    __shared__ float s1[256];
    float sum = 0.0f, sq = 0.0f;
    for (int i = blockIdx.x * blockDim.x + threadIdx.x; i < n; i += gridDim.x * blockDim.x) {
        const float v = agg[i];
        sum += v;
        sq  += v * v;
    }
    s0[threadIdx.x] = sum;
    s1[threadIdx.x] = sq;
    __syncthreads();
    for (int s = 128; s > 0; s >>= 1) {
        if ((int)threadIdx.x < s) {
            s0[threadIdx.x] += s0[threadIdx.x + s];
            s1[threadIdx.x] += s1[threadIdx.x + s];
        }
        __syncthreads();
    }
    if (threadIdx.x == 0) {
        atomicAdd(&red[0], s0[0]);
        atomicAdd(&red[1], s1[0]);
    }
}

// ---------------------------------------------------------------------------
// Normalize + affine + ReLU:  out = relu(lw[c]*(agg-mean)*rstd + lb[c])
// ---------------------------------------------------------------------------
__global__ void __launch_bounds__(256)
k_norm_relu(const float* __restrict__ agg, const float* __restrict__ red,
            const float* __restrict__ lw, const float* __restrict__ lb,
            float* __restrict__ out, int n, float invM) {
    const float mean = red[0] * invM;
    const float var  = red[1] * invM - mean * mean;
    const float rstd = rsqrtf(var + LN_EPS);
    for (int i = blockIdx.x * blockDim.x + threadIdx.x; i < n; i += gridDim.x * blockDim.x) {
        const int c = i & (D - 1);
        const float v = lw[c] * ((agg[i] - mean) * rstd) + lb[c];
        out[i] = v > 0.0f ? v : 0.0f;
    }
}

// ---------------------------------------------------------------------------
// launcher
// ---------------------------------------------------------------------------
extern "C" void kernel_launch(void* const* d_in, const int* in_sizes, int n_in,
                              void* d_out, int out_size, void* d_ws, size_t ws_size,
                              hipStream_t stream) {
    const float* x    = (const float*)d_in[0];   // [N,64]
    const int*   ei   = (const int*)d_in[1];     // [2,E]
    const float* Ws   = (const float*)d_in[2];   // [L,64,64]
    const float* bs   = (const float*)d_in[3];   // [L,64]
    const float* ln_w = (const float*)d_in[4];   // [L,64]
    const float* ln_b = (const float*)d_in[5];   // [L,64]
    float* outp = (float*)d_out;

    const int N  = in_sizes[0] / D;
    const int E  = in_sizes[1] / 2;
    const int ND = N * D;

    // workspace layout (floats)
    float* ws   = (float*)d_ws;
    float* bufA = ws;                     // N*D
    float* bufB = ws + (size_t)ND;        // N*D
    float* agg  = ws + 2 * (size_t)ND;    // N*D
    float* dinv = ws + 3 * (size_t)ND;    // N   (holds degree first, then rsqrt)
    float* red  = dinv + N;               // 2

    const int thr = 256;
    const int gN   = (N + thr - 1) / thr;
    const int gE   = min(4096, (E + thr - 1) / thr);
    const int gND  = min(4096, (ND + thr - 1) / thr);
    const int gGemm = (N / 16 + 3) / 4 + 1;   // 4 waves/block, 16 rows/wave

    // degree -> dinv (once)
    k_zero_f32<<<gN, thr, 0, stream>>>(dinv, N);
    k_count_deg<<<gE, thr, 0, stream>>>(ei + E, dinv, E);
    k_make_dinv<<<gN, thr, 0, stream>>>(dinv, N);

    const float invM = 1.0f / (float)ND;

    const float* hin = x;
    for (int l = 0; l < NLAYERS; ++l) {
        float* hw = (l & 1) ? bufB : bufA;                       // A,B,A,B
        float* ho = (l == NLAYERS - 1) ? outp : hw;              // norm output reuses hw buffer
        k_gemm_wmma<<<gGemm, 128, 0, stream>>>(hin, Ws + (size_t)l * D * D,
                                               bs + (size_t)l * D, dinv, hw, agg, N);
        k_scatter_edges<<<4096, thr, 0, stream>>>(ei, hw, dinv, agg, E);
        k_zero2<<<1, 1, 0, stream>>>(red);
        k_reduce_stats<<<gND, thr, 0, stream>>>(agg, red, ND);
        k_norm_relu<<<gND, thr, 0, stream>>>(agg, red, ln_w + (size_t)l * D,
                                             ln_b + (size_t)l * D, ho, ND, invM);
        hin = ho;
    }
}